// VideoTextAttention_33036888441453
// MI455X (gfx1250) — compile-verified
//
#include <hip/hip_runtime.h>
#include <hip/hip_bf16.h>

// ---------------------------------------------------------------------------
// Types for CDNA5 WMMA f32 16x16x4 (A: 16x4 = 2 VGPR, B: 4x16 = 2 VGPR,
// C/D: 16x16 f32 = 8 VGPR).
// ---------------------------------------------------------------------------
typedef float v2f __attribute__((ext_vector_type(2)));
typedef float v8f __attribute__((ext_vector_type(8)));

// Fast hardware tanh on gfx1250 (v_tanh_f32) when the builtin exists.
#if __has_builtin(__builtin_amdgcn_tanhf)
#define FAST_TANH(x) __builtin_amdgcn_tanhf(x)
#elif __has_builtin(__builtin_amdgcn_tanh_f32)
#define FAST_TANH(x) __builtin_amdgcn_tanh_f32(x)
#else
#define FAST_TANH(x) tanhf(x)
#endif

__device__ __forceinline__ float wave_sum32(float v) {
#pragma unroll
  for (int o = 16; o > 0; o >>= 1) v += __shfl_xor(v, o, 32);
  return v;
}
__device__ __forceinline__ float wave_max32(float v) {
#pragma unroll
  for (int o = 16; o > 0; o >>= 1) v = fmaxf(v, __shfl_xor(v, o, 32));
  return v;
}

// ---------------------------------------------------------------------------
// Row-major f32 WMMA GEMM: C[M,N] = A[M,K] * B[K,N] (+ bias[n]).
// One wave computes a 32x64 strip: 8 v8f accumulators; each K-step (K+=4)
// loads 2 A fragments (b64 each) + 4 B fragments (8 b32) and issues
// 8 v_wmma_f32_16x16x4_f32 -- B fragments are reused across the two M
// sub-tiles, doubling WMMA density per load.
// Addressing: uniform base pointers Ak/Bk advanced with scalar adds; all
// lane-varying address math is K-invariant 32-bit offsets precomputed before
// the loop (GVS-mode loads, no per-iteration 64-bit VALU address math).
// M % 32 == 0 and K % 4 == 0 guaranteed by caller; ragged N handled by
// clamping the load offset to a valid column and zeroing via v_cndmask.
// tilesN = ceil(N/64).
// ---------------------------------------------------------------------------
__global__ __launch_bounds__(128) void wmma_gemm_f32_kernel(
    const float* __restrict__ A, const float* __restrict__ B,
    const float* __restrict__ bias, float* __restrict__ C,
    int M, int N, int K, int tilesN) {
  const int wave = blockIdx.x * (blockDim.x >> 5) + (threadIdx.x >> 5);
  const int lane = threadIdx.x & 31;
  const int tilesM = M >> 5;  // 32-row strips
  if (wave >= tilesM * tilesN) return;  // wave-uniform: EXEC stays all-ones

  const int tm = wave / tilesN;
  const int tn = wave % tilesN;
  const int row0 = tm << 5;
  const int col0 = tn << 6;

  // ISA fragment mapping (CDNA5 ISA):
  //   A lane -> m = lane&15, holds k = 2*(lane>>4) + {0,1}
  //   B lane -> n = lane&15, holds k = 2*(lane>>4) + {0,1}
  const int m = lane & 15;
  const int kb = (lane >> 4) << 1;
  const int n = lane & 15;

  // K-invariant, lane-varying 32-bit offsets (elements).
  const int aoff0 = (row0 + m) * K + kb;
  const int aoff1 = aoff0 + 16 * K;
  bool cok[4];
  int boff[4];
#pragma unroll
  for (int t = 0; t < 4; ++t) {
    const int c = col0 + 16 * t + n;
    cok[t] = (c < N);
    boff[t] = kb * N + (cok[t] ? c : (N - 1));  // clamped: always in-bounds
  }

  __builtin_prefetch(A + aoff0, 0, 1);
  __builtin_prefetch(B + boff[0], 0, 1);

  v8f acc[8] = {};

  const float* __restrict__ Ak = A;  // uniform, += 4 per step (scalar add)
  const float* __restrict__ Bk = B;  // uniform, += 4*N per step (scalar add)
  const int bstep = N << 2;

  for (int k0 = 0; k0 < K; k0 += 4) {
    v2f a0, a1;
    a0.x = Ak[aoff0];
    a0.y = Ak[aoff0 + 1];
    a1.x = Ak[aoff1];
    a1.y = Ak[aoff1 + 1];

    v2f bf[4];
#pragma unroll
    for (int t = 0; t < 4; ++t) {
      const float x0 = Bk[boff[t]];      // unconditional, address valid
      const float x1 = Bk[boff[t] + N];  // next K row
      bf[t].x = cok[t] ? x0 : 0.0f;      // v_cndmask, no exec branching
      bf[t].y = cok[t] ? x1 : 0.0f;
    }
#pragma unroll
    for (int t = 0; t < 4; ++t) {
      acc[t] = __builtin_amdgcn_wmma_f32_16x16x4_f32(
          false, a0, false, bf[t], (short)0, acc[t], false, false);
    }
#pragma unroll
    for (int t = 0; t < 4; ++t) {
      acc[4 + t] = __builtin_amdgcn_wmma_f32_16x16x4_f32(
          false, a1, false, bf[t], (short)0, acc[4 + t], false, false);
    }
    Ak += 4;
    Bk += (size_t)bstep;
  }

  // C/D layout: lanes 0-15 -> rows 0..7 in v0..v7; lanes 16-31 -> rows 8..15.
  const int rhalf = lane >> 4;
#pragma unroll
  for (int h = 0; h < 2; ++h) {
    const int rowb = row0 + 16 * h;
#pragma unroll
    for (int t = 0; t < 4; ++t) {
      const int coln = col0 + 16 * t + n;
      if (coln < N) {
        const float bv = bias ? bias[coln] : 0.0f;
#pragma unroll
        for (int r = 0; r < 8; ++r) {
          C[(size_t)(rowb + r + 8 * rhalf) * N + coln] = acc[4 * h + t][r] + bv;
        }
      }
    }
  }
}

// ---------------------------------------------------------------------------
// Fused additive-attention core. Per block: one (b, chunk-of-8-p).
// h2[b] (64x512 f32 = 128 KB) + bias + w3 + per-wave score rows live in LDS
// (137 KB -- CDNA5's 320 KB/WGP makes this possible). Each wave owns one p:
//   e[q]   = sum_a tanh(h1[p,a] + h2[q,a] + bias[a]) * w3[a]   (never
//            materializing the 268 MB h tensor)
//   s      = softmax_q(e)
//   ta[a]  = sum_q s[q] * h2[q,a]
//   t[p,:] = tanh(concat(h1, ta, h1*ta, h1-ta))   -> workspace for final GEMM
// Lane owns a-indices {lane, lane+32, ..., lane+480} (16 regs per vector).
// ---------------------------------------------------------------------------
__global__ __launch_bounds__(256) void attn_fuse_kernel(
    const float* __restrict__ h1g, const float* __restrict__ h2g,
    const float* __restrict__ bias, const float* __restrict__ w3,
    float* __restrict__ tbuf) {
  extern __shared__ float smem[];
  float* h2s = smem;                  // 64*512
  float* biass = smem + 64 * 512;     // 512
  float* w3s = biass + 512;           // 512
  float* esc = w3s + 512;             // 8 waves * 64

  const int tid = threadIdx.x;
  const int lane = tid & 31;
  const int w = tid >> 5;
  const int b = blockIdx.x >> 5;
  const int pchunk = blockIdx.x & 31;
  const int p = pchunk * 8 + w;

  const float* __restrict__ h2b = h2g + (size_t)b * (64 * 512);
  for (int i = tid; i < 64 * 512; i += 256) h2s[i] = h2b[i];
  for (int i = tid; i < 512; i += 256) {
    biass[i] = bias[i];
    w3s[i] = w3[i];
  }
  __syncthreads();

  float w3r[16], h1r[16], h1b[16];
#pragma unroll
  for (int j = 0; j < 16; ++j) w3r[j] = w3s[lane + 32 * j];

  const float* __restrict__ h1row = h1g + (size_t)(b * 256 + p) * 512;
#pragma unroll
  for (int j = 0; j < 16; ++j) {
    h1r[j] = h1row[lane + 32 * j];
    h1b[j] = h1r[j] + biass[lane + 32 * j];
  }

  // ---- scores e[q] (trans-pipe bound: 64*512 tanh per p) ----
  float* escw = esc + w * 64;
  for (int q = 0; q < 64; ++q) {
    const float* __restrict__ h2q = h2s + q * 512;
    float acc = 0.0f;
#pragma unroll
    for (int j = 0; j < 16; ++j) {
      const float t = FAST_TANH(h1b[j] + h2q[lane + 32 * j]);
      acc = __builtin_fmaf(t, w3r[j], acc);
    }
    acc = wave_sum32(acc);
    if (lane == 0) escw[q] = acc;
  }

  // ---- softmax over q (64 values, 2 per lane) ----
  const float e0 = escw[lane];
  const float e1 = escw[lane + 32];
  const float mx = wave_max32(fmaxf(e0, e1));
  const float x0 = __expf(e0 - mx);
  const float x1 = __expf(e1 - mx);
  const float inv = 1.0f / wave_sum32(x0 + x1);
  escw[lane] = x0 * inv;
  escw[lane + 32] = x1 * inv;

  // ---- text_attn = scores @ h2 ----
  float ta[16];
#pragma unroll
  for (int j = 0; j < 16; ++j) ta[j] = 0.0f;
  for (int q = 0; q < 64; ++q) {
    const float sq = escw[q];  // LDS broadcast
    const float* __restrict__ h2q = h2s + q * 512;
#pragma unroll
    for (int j = 0; j < 16; ++j)
      ta[j] = __builtin_fmaf(sq, h2q[lane + 32 * j], ta[j]);
  }

  // ---- t[p,:] = tanh(concat(h1, ta, h1*ta, h1-ta)) ----
  float* __restrict__ trow = tbuf + (size_t)(b * 256 + p) * 2048;
#pragma unroll
  for (int j = 0; j < 16; ++j) {
    const int a = lane + 32 * j;
    const float hv = h1r[j];
    const float tv = ta[j];
    trow[a] = FAST_TANH(hv);
    trow[512 + a] = FAST_TANH(tv);
    trow[1024 + a] = FAST_TANH(hv * tv);
    trow[1536 + a] = FAST_TANH(hv - tv);
  }
}

// ---------------------------------------------------------------------------
// Launch: h1-GEMM, h2-GEMM, fused attention, final GEMM(+bias).
// Workspace: h1 (4 MB) | h2 (1 MB) | tanh(cont) (16 MB)  => 21 MB.
// ---------------------------------------------------------------------------
extern "C" void kernel_launch(void* const* d_in, const int* in_sizes, int n_in,
                              void* d_out, int out_size, void* d_ws,
                              size_t ws_size, hipStream_t stream) {
  (void)in_sizes; (void)n_in; (void)out_size; (void)ws_size;
  const float* video = (const float*)d_in[0];  // [8,256,1024]
  const float* text = (const float*)d_in[1];   // [8,64,768]
  // d_in[2], d_in[3]: masks -- provably dead in the reference.
  const float* w1 = (const float*)d_in[4];     // [1024,512]
  const float* w2 = (const float*)d_in[5];     // [768,512]
  const float* w3 = (const float*)d_in[6];     // [512,1]
  const float* bias = (const float*)d_in[7];   // [512]
  const float* w4 = (const float*)d_in[8];     // [2048,500]
  const float* b4 = (const float*)d_in[9];     // [500]
  float* out = (float*)d_out;                  // [8,256,500]

  float* h1 = (float*)d_ws;            // 2048*512
  float* h2 = h1 + 2048 * 512;         // 512*512
  float* tb = h2 + 512 * 512;          // 2048*2048

  // h1 = video @ w1 : M=2048, N=512, K=1024 -> 64*8 = 512 waves -> 128 blocks
  wmma_gemm_f32_kernel<<<128, 128, 0, stream>>>(video, w1, nullptr, h1,
                                                2048, 512, 1024, 8);
  // h2 = text @ w2 : M=512, N=512, K=768 -> 16*8 = 128 waves -> 32 blocks
  wmma_gemm_f32_kernel<<<32, 128, 0, stream>>>(text, w2, nullptr, h2,
                                               512, 512, 768, 8);

  const size_t smem = (size_t)(64 * 512 + 512 + 512 + 8 * 64) * sizeof(float);
  hipFuncSetAttribute((const void*)attn_fuse_kernel,
                      hipFuncAttributeMaxDynamicSharedMemorySize, (int)smem);
  attn_fuse_kernel<<<256, 256, smem, stream>>>(h1, h2, bias, w3, tb);

  // out = tanh(cont) @ w4 + b4 : M=2048, N=500, K=2048 -> 512 waves
  wmma_gemm_f32_kernel<<<128, 128, 0, stream>>>(tb, w4, b4, out,
                                                2048, 500, 2048, 8);
}